// GLREModule_35759897706775
// MI455X (gfx1250) — compile-verified
//
#include <hip/hip_runtime.h>

#define DEVI __device__ __forceinline__

namespace {
constexpr int B = 8, N = 224, E = 64, M = 128, D = 256, IN = 300, R = 5, H = 4;
constexpr int P = E * E;            // 4096 entity pairs
constexpr int OUTROWS = N + 2 * P;  // 8416 rows per batch in d_out
constexpr int KC = 32;              // K-chunk staged in LDS per iteration
constexpr int APAD = 36;            // padded A row stride in LDS (floats)
}

typedef __attribute__((ext_vector_type(2))) float v2f;
typedef __attribute__((ext_vector_type(8))) float v8f;

// ---------------------------------------------------------------------------
// CDNA5 f32 WMMA: D(16x16,f32) = A(16x4,f32) * B(4x16,f32) + C
// A frag: lanes 0-15 row=lane (K0,K1), lanes 16-31 row=lane-16 (K2,K3).
// B frag: lanes 0-15 col=lane (K0,K1), lanes 16-31 col=lane-16 (K2,K3).
// C/D:    VGPR v -> row v (lanes 0-15) / row v+8 (lanes 16-31), col=lane&15.
// ---------------------------------------------------------------------------
DEVI v8f wmma4(v2f a, v2f b, v8f c) {
  return __builtin_amdgcn_wmma_f32_16x16x4_f32(false, a, false, b, (short)0, c,
                                               false, false);
}

// ---------------------------------------------------------------------------
// LDS-staged block GEMM core. Block = 128 threads (4 waves):
//   block tile 64(M) x 64(N), K consumed in KC=32 chunks, double-buffered.
// lA:  [buf][wave][16 x APAD]  row-major A chunk (pad => conflict-free b64)
// lBf: [buf][ks][tile][lane]   B fragments pre-swizzled: one ds_load_b64 per
//                              wmma operand, 32 lanes x 8B = 64 banks exactly.
// ---------------------------------------------------------------------------
struct TileLds {
  float lA[2][4][16 * APAD];
  float2 lBf[2][KC / 4][4][32];
};

// Stage one KC chunk cooperatively (all 128 threads). Guards handle the
// K remainder (zeros) and partial-M blocks (zeros).
DEVI void stage_chunk(TileLds& S, int buf, const float* __restrict__ Ab,
                      int lda, int Mvalid, const float* __restrict__ Bb,
                      int ldb, int k0, int K, int tid) {
  const float4 z4 = make_float4(0.f, 0.f, 0.f, 0.f);
  // ---- A: 64 rows x 32 k  (512 float4, 4 per thread, coalesced) ----------
#pragma unroll
  for (int it = 0; it < 4; ++it) {
    int idx = tid + it * 128;
    int row = idx >> 3;      // 0..63
    int k4 = idx & 7;        // float4 index along k
    float4 v = z4;
    if (row < Mvalid && (k0 + k4 * 4) < K)
      v = *(const float4*)(Ab + (size_t)row * lda + k0 + k4 * 4);
    float* d = &S.lA[buf][row >> 4][(row & 15) * APAD + k4 * 4];
    ((float2*)d)[0] = make_float2(v.x, v.y);
    ((float2*)d)[1] = make_float2(v.z, v.w);
  }
  // ---- B: 32 k x 64 n (512 float4, 4 per thread) -> fragment swizzle -----
#pragma unroll
  for (int it = 0; it < 4; ++it) {
    int idx = tid + it * 128;
    int kk = idx >> 4;       // 0..31
    int c4 = idx & 15;       // float4 index along n (c = c4*4)
    float4 v = z4;
    if ((k0 + kk) < K)
      v = *(const float4*)(Bb + (size_t)(k0 + kk) * ldb + c4 * 4);
    int ks = kk >> 2;
    int koff2 = (kk >> 1) & 1;
    int vi = kk & 1;
    int t = c4 >> 2;
    int l = ((c4 & 3) << 2) + (koff2 << 4);
    float* d = (float*)&S.lBf[buf][ks][t][l];
    d[0 + vi] = v.x;
    d[2 + vi] = v.y;
    d[4 + vi] = v.z;
    d[6 + vi] = v.w;
  }
}

// 8 k-steps (KC/4) of WMMA from LDS for one wave's 16x64 tile.
DEVI void compute_chunk(TileLds& S, int buf, int wave, int lane, v8f acc[4]) {
  const int r = lane & 15;
  const int koff = (lane >> 4) << 1;
  const float* lAw = S.lA[buf][wave];
#pragma unroll
  for (int ks = 0; ks < KC / 4; ++ks) {
    float2 ap = *(const float2*)&lAw[r * APAD + ks * 4 + koff];
    v2f a;
    a.x = ap.x;
    a.y = ap.y;
#pragma unroll
    for (int t = 0; t < 4; ++t) {
      float2 bp = S.lBf[buf][ks][t][lane];
      v2f bv;
      bv.x = bp.x;
      bv.y = bp.y;
      acc[t] = wmma4(a, bv, acc[t]);
    }
  }
}

// ---------------------------------------------------------------------------
// Generic batched GEMM: C = A @ B (+bias). Batch z: zo=z/innerB, zi=z%innerB.
// grid = (Ncols/64, ceil(M/64), batches), block = 128.
// ---------------------------------------------------------------------------
__global__ void gemm_kernel(const float* __restrict__ A,
                            const float* __restrict__ Bm,
                            const float* __restrict__ bias,
                            float* __restrict__ C, int Mrows, int K, int lda,
                            int ldb, int ldc, int innerB, long long sAo,
                            long long sAi, long long sBo, long long sBi,
                            long long sBiasi, long long sCo, long long sCi) {
  __shared__ TileLds S;
  const int z = blockIdx.z;
  const int zo = z / innerB, zi = z % innerB;
  const int tid = threadIdx.x;
  const int wave = tid >> 5, lane = tid & 31;
  const int m0blk = blockIdx.y * 64;
  const int m0 = m0blk + wave * 16;
  const int n0 = blockIdx.x * 64;
  const float* Ab = A + zo * sAo + zi * sAi + (size_t)m0blk * lda;
  const float* Bb = Bm + zo * sBo + zi * sBi + n0;
  const int Mvalid = Mrows - m0blk;  // rows valid in this block (>0)
  const int NCHUNK = (K + KC - 1) / KC;

  stage_chunk(S, 0, Ab, lda, Mvalid, Bb, ldb, 0, K, tid);
  __syncthreads();

  v8f zf = {0, 0, 0, 0, 0, 0, 0, 0};
  v8f acc[4] = {zf, zf, zf, zf};
  int buf = 0;
  for (int ch = 0; ch < NCHUNK; ++ch) {
    if (ch + 1 < NCHUNK) {
      if (ch + 2 < NCHUNK)  // hint L2 for the chunk after next
        __builtin_prefetch(Bb + (size_t)(ch + 2) * KC * ldb, 0, 1);
      stage_chunk(S, buf ^ 1, Ab, lda, Mvalid, Bb, ldb, (ch + 1) * KC, K, tid);
    }
    if (m0 < Mrows) compute_chunk(S, buf, wave, lane, acc);
    if (ch + 1 < NCHUNK) {
      __syncthreads();
      buf ^= 1;
    }
  }

  if (m0 < Mrows) {
    float* Cb = C + zo * sCo + zi * sCi + (size_t)m0 * ldc + n0;
    const int colb = lane & 15;
    const int rofs = (lane >> 4) << 3;
    const float* biasb = bias ? (bias + zi * sBiasi + n0) : nullptr;
#pragma unroll
    for (int t = 0; t < 4; ++t) {
#pragma unroll
      for (int v = 0; v < 8; ++v) {
        float val = acc[t][v];
        int col = t * 16 + colb;
        if (biasb) val += biasb[col];
        Cb[(size_t)(rofs + v) * ldc + col] = val;
      }
    }
  }
}

// ---------------------------------------------------------------------------
// RGCN combine: Out[b,n,d] = relu((sum_r adj[b,r]@xW[b,r] + g[b,n,d]) / den)
// Same LDS pipeline; chunk index runs over (r, k-chunk). N=224 = 7*KC exact.
// ---------------------------------------------------------------------------
__global__ void rgcn_combine_kernel(const float* __restrict__ adj,
                                    const float* __restrict__ xW,
                                    const float* __restrict__ g,
                                    const float* __restrict__ den,
                                    float* __restrict__ Out,
                                    long long outBatchStride) {
  __shared__ TileLds S;
  const int b = blockIdx.z;
  const int tid = threadIdx.x;
  const int wave = tid >> 5, lane = tid & 31;
  const int m0blk = blockIdx.y * 64;
  const int m0 = m0blk + wave * 16;
  const int n0 = blockIdx.x * 64;
  const int Mvalid = N - m0blk;
  constexpr int CPR = N / KC;  // 7 chunks per relation
  constexpr int NCHUNK = R * CPR;

  auto Aof = [&](int ch) {
    int r = ch / CPR;
    return adj + (((size_t)b * R + r) * N + m0blk) * N;
  };
  auto Bof = [&](int ch) {
    int r = ch / CPR;
    return xW + ((size_t)(b * R + r) * N) * D + n0;
  };
  auto k0of = [&](int ch) { return (ch % CPR) * KC; };

  stage_chunk(S, 0, Aof(0), N, Mvalid, Bof(0), D, k0of(0), N, tid);
  __syncthreads();

  v8f zf = {0, 0, 0, 0, 0, 0, 0, 0};
  v8f acc[4] = {zf, zf, zf, zf};
  int buf = 0;
  for (int ch = 0; ch < NCHUNK; ++ch) {
    if (ch + 1 < NCHUNK)
      stage_chunk(S, buf ^ 1, Aof(ch + 1), N, Mvalid, Bof(ch + 1), D,
                  k0of(ch + 1), N, tid);
    if (m0 < N) compute_chunk(S, buf, wave, lane, acc);
    if (ch + 1 < NCHUNK) {
      __syncthreads();
      buf ^= 1;
    }
  }

  if (m0 < N) {
    const int colb = lane & 15;
    const int rofs = (lane >> 4) << 3;
#pragma unroll
    for (int t = 0; t < 4; ++t) {
#pragma unroll
      for (int v = 0; v < 8; ++v) {
        int row = m0 + rofs + v;
        int col = n0 + t * 16 + colb;
        float val = (acc[t][v] + g[((size_t)b * N + row) * D + col]) /
                    den[b * N + row];
        Out[(size_t)b * outBatchStride + (size_t)row * D + col] =
            fmaxf(val, 0.0f);
      }
    }
  }
}

// ---------------------------------------------------------------------------
// denom[b,n] = 1 + sum_{r,m} adj[b,r,n,m]   (one wave per (b,n))
// ---------------------------------------------------------------------------
__global__ void denom_kernel(const float* __restrict__ adj,
                             float* __restrict__ den) {
  const int w = blockIdx.x * (blockDim.x >> 5) + (threadIdx.x >> 5);
  const int lane = threadIdx.x & 31;
  if (w >= B * N) return;
  const int b = w / N, n = w % N;
  float s = 0.f;
  for (int r = 0; r < R; ++r) {
    const float* row = adj + (((size_t)b * R + r) * N + n) * N;
    for (int m = lane; m < N; m += 32) s += row[m];
  }
  for (int off = 16; off; off >>= 1) s += __shfl_xor(s, off, 32);
  if (lane == 0) den[w] = s + 1.0f;
}

// mention_sen_rep[b,m,:] = sentences[info_sid[b,m], :]
__global__ void gather_msr_kernel(const float* __restrict__ sentences,
                                  const int* __restrict__ sid,
                                  float* __restrict__ msr) {
  const int t = blockIdx.x * blockDim.x + threadIdx.x;  // B*M*(D/4)
  const int c = t % (D / 4);
  const int r = t / (D / 4);
  if (r >= B * M) return;
  const int s = sid[r];
  ((float4*)msr)[(size_t)r * (D / 4) + c] =
      ((const float4*)sentences)[(size_t)s * (D / 4) + c];
}

// s0[b,h,q,m] = <Qe[b,q,h*64:], K[b,m,h*64:]> / 8    (queries are the E
// distinct entities: pre-mask scores depend only on the query entity)
__global__ void score_kernel(const float* __restrict__ Qe,
                             const float* __restrict__ Km,
                             float* __restrict__ s0) {
  const int t = blockIdx.x * blockDim.x + threadIdx.x;
  if (t >= B * H * E * M) return;
  int m = t % M;
  int rest = t / M;
  int q = rest % E;
  rest /= E;
  int h = rest % H;
  int b = rest / H;
  const float4* qv = (const float4*)(Qe + ((size_t)b * E + q) * D + h * 64);
  const float4* kv = (const float4*)(Km + ((size_t)b * M + m) * D + h * 64);
  float acc = 0.f;
#pragma unroll
  for (int c = 0; c < 16; ++c) {
    float4 a = qv[c], x = kv[c];
    acc += a.x * x.x + a.y * x.y + a.z * x.z + a.w * x.w;
  }
  s0[t] = acc * 0.125f;  // 1/sqrt(dk), dk=64
}

// ---------------------------------------------------------------------------
// Masked softmax + attention@V. One wave per output row (b, pair p).
// mode_h=1: query=j, mask-entity=i ; mode_h=0: query=i, mask-entity=j.
// ---------------------------------------------------------------------------
__global__ void attn_ctx_kernel(const float* __restrict__ s0,
                                const float* __restrict__ V,
                                const int* __restrict__ eid,
                                float* __restrict__ ctx, int mode_h) {
  const int w = blockIdx.x * (blockDim.x >> 5) + (threadIdx.x >> 5);
  const int lane = threadIdx.x & 31;
  if (w >= B * P) return;
  const int b = w / P;
  const int p = w % P;
  const int i = p >> 6;  // p / E
  const int j = p & 63;  // p % E
  const int q = mode_h ? j : i;
  const int u = mode_h ? i : j;
  for (int h = 0; h < H; ++h) {
    const float* srow = s0 + ((size_t)(b * H + h) * E + q) * M;
    float val[4];
#pragma unroll
    for (int c = 0; c < 4; ++c) {
      int m = c * 32 + lane;
      float s = srow[m];
      val[c] = (eid[b * M + m] != u) ? -1.0e9f : s;
    }
    float mx = fmaxf(fmaxf(val[0], val[1]), fmaxf(val[2], val[3]));
    for (int off = 16; off; off >>= 1) mx = fmaxf(mx, __shfl_xor(mx, off, 32));
    float sum = 0.f;
#pragma unroll
    for (int c = 0; c < 4; ++c) {
      val[c] = __expf(val[c] - mx);
      sum += val[c];
    }
    for (int off = 16; off; off >>= 1) sum += __shfl_xor(sum, off, 32);
    const float inv = 1.0f / sum;
    float c0 = 0.f, c1 = 0.f;
    const float* Vh = V + (size_t)b * M * D + h * 64;
#pragma unroll 1
    for (int c = 0; c < 4; ++c) {
      for (int sl = 0; sl < 32; ++sl) {
        float am = __shfl(val[c], sl, 32);  // wave-uniform value
        if (am != 0.0f) {                   // masked weights underflow to 0
          int m = c * 32 + sl;
          c0 += am * Vh[(size_t)m * D + lane];
          c1 += am * Vh[(size_t)m * D + 32 + lane];
        }
      }
    }
    size_t o = ((size_t)b * P + p) * D + h * 64;
    ctx[o + lane] = c0 * inv;
    ctx[o + 32 + lane] = c1 * inv;
  }
}

// ---------------------------------------------------------------------------
extern "C" void kernel_launch(void* const* d_in, const int* in_sizes, int n_in,
                              void* d_out, int out_size, void* d_ws,
                              size_t ws_size, hipStream_t stream) {
  (void)in_sizes; (void)n_in; (void)out_size; (void)ws_size;
  const float* nodes = (const float*)d_in[0];
  const float* adj = (const float*)d_in[1];
  const float* mentions = (const float*)d_in[2];
  const float* sentences = (const float*)d_in[3];
  const float* Wr0 = (const float*)d_in[4];
  const float* br0 = (const float*)d_in[5];
  const float* Wr1 = (const float*)d_in[6];
  const float* br1 = (const float*)d_in[7];
  const float* W00 = (const float*)d_in[8];
  const float* b00 = (const float*)d_in[9];
  const float* W01 = (const float*)d_in[10];
  const float* b01 = (const float*)d_in[11];
  const float* Wq[2] = {(const float*)d_in[12], (const float*)d_in[16]};
  const float* Wk[2] = {(const float*)d_in[13], (const float*)d_in[17]};
  const float* Wv[2] = {(const float*)d_in[14], (const float*)d_in[18]};
  const float* Wo[2] = {(const float*)d_in[15], (const float*)d_in[19]};
  const int* eid = (const int*)d_in[20];
  const int* sid = (const int*)d_in[21];
  float* out = (float*)d_out;
  float* ws = (float*)d_ws;

  // workspace layout (floats)
  size_t o = 0;
  float* den = ws + o;  o += 2048;                   // B*N rounded up
  float* xW = ws + o;   o += (size_t)B * R * N * D;  // per-relation xW
  float* g = ws + o;    o += (size_t)B * N * D;      // x@W0 + b0
  float* hbuf = ws + o; o += (size_t)B * N * D;      // layer-1 output
  float* msr = ws + o;  o += (size_t)B * M * D;      // gathered sentences
  float* Qe[2] = {ws + o, ws + o + (size_t)B * E * D};
  o += 2 * (size_t)B * E * D;
  float* Km[2] = {ws + o, ws + o + (size_t)B * M * D};
  o += 2 * (size_t)B * M * D;
  float* Vm[2] = {ws + o, ws + o + (size_t)B * M * D};
  o += 2 * (size_t)B * M * D;
  float* s0[2] = {ws + o, ws + o + (size_t)B * H * E * M};
  o += 2 * (size_t)B * H * E * M;
  float* ctx = ws + o;  // B*P*D, reused across both attention modes

  auto gemm = [&](const float* A, const float* Bm, const float* bias, float* C,
                  int Mrows, int K, int lda, int ldb, int ldc, int Ncols,
                  int batches, int innerB, long long sAo, long long sAi,
                  long long sBo, long long sBi, long long sBiasi,
                  long long sCo, long long sCi) {
    dim3 grid(Ncols / 64, (Mrows + 63) / 64, batches);
    gemm_kernel<<<grid, 128, 0, stream>>>(A, Bm, bias, C, Mrows, K, lda, ldb,
                                          ldc, innerB, sAo, sAi, sBo, sBi,
                                          sBiasi, sCo, sCi);
  };

  // ---- denominators -------------------------------------------------------
  denom_kernel<<<(B * N) / 8, 256, 0, stream>>>(adj, den);

  // ---- RGCN layer 1 -------------------------------------------------------
  gemm(nodes, Wr0, br0, xW, N, IN, IN, D, D, D, B * R, R, (long long)N * IN, 0,
       0, (long long)IN * D, D, (long long)R * N * D, (long long)N * D);
  gemm(nodes, W00, b00, g, N, IN, IN, D, D, D, B, 1, (long long)N * IN, 0, 0,
       0, 0, (long long)N * D, 0);
  rgcn_combine_kernel<<<dim3(D / 64, (N + 63) / 64, B), 128, 0, stream>>>(
      adj, xW, g, den, hbuf, (long long)N * D);

  // ---- RGCN layer 2 (writes gcn_out straight into d_out rows [0,N)) -------
  gemm(hbuf, Wr1, br1, xW, N, D, D, D, D, D, B * R, R, (long long)N * D, 0, 0,
       (long long)D * D, D, (long long)R * N * D, (long long)N * D);
  gemm(hbuf, W01, b01, g, N, D, D, D, D, D, B, 1, (long long)N * D, 0, 0, 0, 0,
       (long long)N * D, 0);
  rgcn_combine_kernel<<<dim3(D / 64, (N + 63) / 64, B), 128, 0, stream>>>(
      adj, xW, g, den, out, (long long)OUTROWS * D);

  // ---- attention inputs ---------------------------------------------------
  gather_msr_kernel<<<(B * M * (D / 4) + 255) / 256, 256, 0, stream>>>(
      sentences, sid, msr);

  for (int md = 0; md < 2; ++md) {  // md=0 -> h-head, md=1 -> t-head
    // Qe = ents @ Wq  (ents = first E rows of each batch of d_out)
    gemm(out, Wq[md], nullptr, Qe[md], E, D, D, D, D, D, B, 1,
         (long long)OUTROWS * D, 0, 0, 0, 0, (long long)E * D, 0);
    // K = mention_sen_rep @ Wk ; V = mentions @ Wv
    gemm(msr, Wk[md], nullptr, Km[md], M, D, D, D, D, D, B, 1,
         (long long)M * D, 0, 0, 0, 0, (long long)M * D, 0);
    gemm(mentions, Wv[md], nullptr, Vm[md], M, D, D, D, D, D, B, 1,
         (long long)M * D, 0, 0, 0, 0, (long long)M * D, 0);
    score_kernel<<<(B * H * E * M + 255) / 256, 256, 0, stream>>>(
        Qe[md], Km[md], s0[md]);
  }

  // ---- per-pair masked softmax + AV, then ctx @ Wo into d_out -------------
  for (int md = 0; md < 2; ++md) {
    attn_ctx_kernel<<<(B * P) / 8, 256, 0, stream>>>(s0[md], Vm[md], eid, ctx,
                                                     md == 0 ? 1 : 0);
    gemm(ctx, Wo[md], nullptr, out + (size_t)(N + md * P) * D, P, D, D, D, D,
         D, B, 1, (long long)P * D, 0, 0, 0, 0, (long long)OUTROWS * D, 0);
  }
}